// PyGRMSNorm_82016695485249
// MI455X (gfx1250) — compile-verified
//
#include <hip/hip_runtime.h>

// ---------------------------------------------------------------------------
// PyG-style graph RMSNorm, collapsed:
//   rms[g]   = sqrt( sumsq[g] / (count[g]*H) + eps )        (per-graph scalar)
//   out[n,h] = weight[h] * x[n,h] / rms[batch[n]]
// Bandwidth-bound: ~3.1 GB traffic -> ~133 us at 23.3 TB/s.
// WMMA f32 16x16x4 (B = ones) is used as the cross-lane row-sum reducer.
// ---------------------------------------------------------------------------

typedef __attribute__((ext_vector_type(2))) float v2f;
typedef __attribute__((ext_vector_type(8))) float v8f;

#define HIDDEN      256
#define NUM_GRAPHS  1024
#define EPS_F       1e-6f

// --------------------------------------------------------------------------
// Zero the per-graph accumulators (must run every call; harness poisons ws).
// --------------------------------------------------------------------------
__global__ void zero_sumsq_kernel(float* __restrict__ sumsq) {
    int i = blockIdx.x * blockDim.x + threadIdx.x;
    if (i < NUM_GRAPHS) sumsq[i] = 0.0f;
}

// --------------------------------------------------------------------------
// Pass 1: per-graph sum of squares.
// One wave owns a tile of 16 rows. Lane layout matches V_WMMA_F32_16X16X4_F32
// A-matrix: lanes 0-15 = (M=lane, K=0), lanes 16-31 = (M=lane-16, K=2).
// Lane (m, half) accumulates sumsq of row m over columns [half*128, half*128+128).
// One WMMA with B = ones reduces the 32 partials into 16 row sums; lanes 0 and
// 16 then atomically add 8 row sums each into sumsq[batch[row]].
// --------------------------------------------------------------------------
__global__ __launch_bounds__(256) void pass1_sumsq_kernel(
        const float* __restrict__ x,
        const int*   __restrict__ batch,
        float*       __restrict__ sumsq,
        int n_nodes) {
    const int lane         = threadIdx.x & 31;
    const int waveInBlock  = threadIdx.x >> 5;
    const int wavesPerBlk  = blockDim.x >> 5;
    const int wave         = blockIdx.x * wavesPerBlk + waveInBlock;
    const int nWaves       = gridDim.x * wavesPerBlk;
    const int nTiles       = (n_nodes + 15) >> 4;

    const int m    = lane & 15;   // row within tile (A-matrix M index)
    const int half = lane >> 4;   // which 128-column half of the row

    for (int tile = wave; tile < nTiles; tile += nWaves) {
        const int rowBase = tile << 4;
        const int row     = rowBase + m;

        float acc = 0.0f;
        if (row < n_nodes) {
            // 32 x float4 = 128 floats = this lane's half-row.
            const float4* p =
                (const float4*)(x + (size_t)row * HIDDEN) + half * 32;
#pragma unroll 8
            for (int j = 0; j < 32; ++j) {
                float4 v = p[j];
                acc = fmaf(v.x, v.x, acc);
                acc = fmaf(v.y, v.y, acc);
                acc = fmaf(v.z, v.z, acc);
                acc = fmaf(v.w, v.w, acc);
            }
        }

        // A[m, K]: K=0 (lanes 0-15) and K=2 (lanes 16-31) carry the partials,
        // K=1/K=3 are zero. B = all-ones => D[m, n] = full row-sum for every n.
        v2f a; a.x = acc;  a.y = 0.0f;
        v2f b; b.x = 1.0f; b.y = 1.0f;
        v8f c = {};
        v8f d = __builtin_amdgcn_wmma_f32_16x16x4_f32(
            /*neg_a=*/false, a, /*neg_b=*/false, b,
            /*c_mod=*/(short)0, c, /*reuse_a=*/false, /*reuse_b=*/false);

        // D layout (32-bit C/D 16x16): VGPR i holds M=i on lanes 0-15 and
        // M=8+i on lanes 16-31 (N = lane column; all columns identical here).
        if (lane == 0 || lane == 16) {
            const int mBase = (lane == 16) ? 8 : 0;
#pragma unroll
            for (int i = 0; i < 8; ++i) {
                const int r = rowBase + mBase + i;
                if (r < n_nodes) {
                    const int g = batch[r];
                    atomicAdd(&sumsq[g], d[i]);
                }
            }
        }
    }
}

// --------------------------------------------------------------------------
// Pass 2: per-graph inverse RMS. Counts recovered from sorted `batch` via
// binary search (no count atomics in pass 1).
// --------------------------------------------------------------------------
__global__ void pass2_invrms_kernel(
        const float* __restrict__ sumsq,
        const int*   __restrict__ batch,
        float*       __restrict__ inv_rms,
        int n_nodes) {
    const int g = blockIdx.x * blockDim.x + threadIdx.x;
    if (g >= NUM_GRAPHS) return;

    auto lower_bound = [&](int key) {
        int lo = 0, hi = n_nodes;
        while (lo < hi) {
            int mid = (lo + hi) >> 1;
            if (batch[mid] < key) lo = mid + 1; else hi = mid;
        }
        return lo;
    };
    int cnt = lower_bound(g + 1) - lower_bound(g);
    if (cnt < 1) cnt = 1;

    const float mean = sumsq[g] / ((float)cnt * (float)HIDDEN);
    inv_rms[g] = 1.0f / sqrtf(mean + EPS_F);
}

// --------------------------------------------------------------------------
// Pass 3: out[n,h] = weight[h] * x[n,h] * inv_rms[batch[n]].
// Perfectly coalesced float4 stream; batch/inv_rms/weight are cache-hot.
// --------------------------------------------------------------------------
__global__ __launch_bounds__(256) void pass3_scale_kernel(
        const float* __restrict__ x,
        const int*   __restrict__ batch,
        const float* __restrict__ weight,
        const float* __restrict__ inv_rms,
        float*       __restrict__ out,
        long long total4) {
    const float4* x4 = (const float4*)x;
    const float4* w4 = (const float4*)weight;
    float4*       o4 = (float4*)out;

    long long idx = (long long)blockIdx.x * blockDim.x + threadIdx.x;
    if (idx >= total4) return;

    const int row  = (int)(idx >> 6);   // 64 float4 per row (HIDDEN/4)
    const int col4 = (int)(idx & 63);

    const float  s = inv_rms[batch[row]];
    const float4 v = x4[idx];
    const float4 w = w4[col4];

    float4 r;
    r.x = w.x * v.x * s;
    r.y = w.y * v.y * s;
    r.z = w.z * v.z * s;
    r.w = w.w * v.w * s;
    o4[idx] = r;
}

// --------------------------------------------------------------------------
// Host-side launcher (graph-capture safe: kernels only, all on `stream`).
// Inputs: d_in[0]=x [N,256] f32, d_in[1]=batch [N] int, d_in[2]=weight [256] f32.
// --------------------------------------------------------------------------
extern "C" void kernel_launch(void* const* d_in, const int* in_sizes, int n_in,
                              void* d_out, int out_size, void* d_ws, size_t ws_size,
                              hipStream_t stream) {
    const float* x      = (const float*)d_in[0];
    const int*   batch  = (const int*)d_in[1];
    const float* weight = (const float*)d_in[2];
    float*       out    = (float*)d_out;

    const int n_nodes = in_sizes[1];     // batch has one entry per node

    float* sumsq   = (float*)d_ws;            // [NUM_GRAPHS]
    float* inv_rms = sumsq + NUM_GRAPHS;      // [NUM_GRAPHS]

    // Zero accumulators every call (deterministic, ws is not re-poisoned).
    zero_sumsq_kernel<<<(NUM_GRAPHS + 255) / 256, 256, 0, stream>>>(sumsq);

    // Pass 1: one wave per 16-row tile (grid-stride).
    const int nTiles  = (n_nodes + 15) / 16;
    const int wavesPB = 256 / 32;
    const int blocks1 = (nTiles + wavesPB - 1) / wavesPB;
    pass1_sumsq_kernel<<<blocks1, 256, 0, stream>>>(x, batch, sumsq, n_nodes);

    // Pass 2: 1024 graphs.
    pass2_invrms_kernel<<<(NUM_GRAPHS + 255) / 256, 256, 0, stream>>>(
        sumsq, batch, inv_rms, n_nodes);

    // Pass 3: one float4 per thread, fully coalesced.
    const long long total4  = (long long)n_nodes * (HIDDEN / 4);
    const long long blocks3 = (total4 + 255) / 256;
    pass3_scale_kernel<<<(unsigned int)blocks3, 256, 0, stream>>>(
        x, batch, weight, inv_rms, out, total4);
}